// MahjongEmbedding_65524021068312
// MI455X (gfx1250) — compile-verified
//
#include <hip/hip_runtime.h>
#include <hip/hip_bf16.h>

#define B_ 512
#define S_ 512
#define D_ 512
#define KDIM 384
#define LN_EPS 1e-5f
#define SENTINEL 224

typedef float v2f __attribute__((ext_vector_type(2)));
typedef float v4f __attribute__((ext_vector_type(4)));
typedef float v8f __attribute__((ext_vector_type(8)));

// ---------------------------------------------------------------------------
// Kernel 1: build concat[B,384] = [scores_e(32) | oya_e(16) | dora_e(320) | hrs_e(16)]
// One block (384 threads = 12 waves) per batch row; each thread owns one element.
// ---------------------------------------------------------------------------
__global__ void build_concat(
    const float* __restrict__ scores,     // [B,4]
    const int*   __restrict__ oya,        // [B]
    const int*   __restrict__ dora,       // [B,5]
    const float* __restrict__ hrs,        // [B,2]
    const float* __restrict__ oya_table,  // [4,16]
    const float* __restrict__ dora_table, // [38,64]
    const float* __restrict__ ln_gamma,   // [4]
    const float* __restrict__ ln_beta,    // [4]
    const float* __restrict__ W_scores,   // [4,32]
    const float* __restrict__ b_scores,   // [32]
    const float* __restrict__ W_hrs,      // [2,16]
    const float* __restrict__ b_hrs,      // [16]
    float* __restrict__ concat)           // [B,384]
{
    const int b = blockIdx.x;
    const int t = threadIdx.x;
    float out;
    if (t < 32) {
        const float x0 = scores[b * 4 + 0];
        const float x1 = scores[b * 4 + 1];
        const float x2 = scores[b * 4 + 2];
        const float x3 = scores[b * 4 + 3];
        const float mu = (x0 + x1 + x2 + x3) * 0.25f;
        const float d0 = x0 - mu, d1 = x1 - mu, d2 = x2 - mu, d3 = x3 - mu;
        const float var = (d0 * d0 + d1 * d1 + d2 * d2 + d3 * d3) * 0.25f;
        const float r = rsqrtf(var + LN_EPS);
        float acc = b_scores[t];
        acc += (d0 * r * ln_gamma[0] + ln_beta[0]) * W_scores[0 * 32 + t];
        acc += (d1 * r * ln_gamma[1] + ln_beta[1]) * W_scores[1 * 32 + t];
        acc += (d2 * r * ln_gamma[2] + ln_beta[2]) * W_scores[2 * 32 + t];
        acc += (d3 * r * ln_gamma[3] + ln_beta[3]) * W_scores[3 * 32 + t];
        out = acc;
    } else if (t < 48) {
        out = oya_table[oya[b] * 16 + (t - 32)];
    } else if (t < 368) {
        const int j = t - 48;
        const int d = j >> 6;      // which dora slot (0..4)
        const int k = j & 63;      // element within 64-wide embedding
        out = dora_table[dora[b * 5 + d] * 64 + k];
    } else {
        const int j = t - 368;
        out = hrs[b * 2 + 0] * W_hrs[j] + hrs[b * 2 + 1] * W_hrs[16 + j] + b_hrs[j];
    }
    concat[b * KDIM + t] = out;
}

// ---------------------------------------------------------------------------
// Kernel 2: embeded_info[512,512] = concat[512,384] @ W_info[384,512] + b_info
// One wave32 per 16x16 output tile, V_WMMA_F32_16X16X4_F32 over K (96 steps).
// A lane layout (ISA 7.12.2, 32-bit A 16x4): lane L holds M=L%16,
//   K = k0 + 2*(L>>4) + {0,1} in the two A VGPRs. B is symmetric (rows of the
//   4x16 K-major B striped across lanes). C/D: VGPR v -> M = v + 8*(L>>4).
// ---------------------------------------------------------------------------
__global__ void gemm_info(
    const float* __restrict__ A,    // concat [512,384]
    const float* __restrict__ W,    // W_info [384,512]
    const float* __restrict__ bias, // [512]
    float* __restrict__ out)        // [512,512]
{
    const int lane = threadIdx.x & 31;
    const int wave = threadIdx.x >> 5;
    const int tile = blockIdx.x * 4 + wave;   // 1024 tiles total
    const int tileM = tile >> 5;              // 0..31
    const int tileN = tile & 31;              // 0..31
    const int m    = lane & 15;
    const int half = lane >> 4;               // 0 or 1 (selects K pair)
    const int arow = tileM * 16 + m;          // A row held by this lane
    const int bcol = tileN * 16 + m;          // B/D column held by this lane

    v8f c = {0.f, 0.f, 0.f, 0.f, 0.f, 0.f, 0.f, 0.f};
    for (int k0 = 0; k0 < KDIM; k0 += 4) {
        const int ka = k0 + 2 * half;
        v2f av, bv;
        av.x = A[arow * KDIM + ka];
        av.y = A[arow * KDIM + ka + 1];
        bv.x = W[ka * D_ + bcol];
        bv.y = W[(ka + 1) * D_ + bcol];
        c = __builtin_amdgcn_wmma_f32_16x16x4_f32(
                /*neg_a=*/false, av, /*neg_b=*/false, bv,
                /*c_mod=*/(short)0, c, /*reuse_a=*/false, /*reuse_b=*/false);
    }
    const float bb = bias[bcol];
#pragma unroll
    for (int v = 0; v < 8; ++v) {
        const int row = tileM * 16 + v + 8 * half;
        out[row * D_ + bcol] = c[v] + bb;
    }
}

// ---------------------------------------------------------------------------
// Sentinel occurrence index machinery (general row-major cumsum semantics).
// ---------------------------------------------------------------------------
__global__ void count_sentinels(const int* __restrict__ action, int* __restrict__ rowCnt) {
    __shared__ int sh[256];
    const int b = blockIdx.x;
    const int t = threadIdx.x;
    int cnt = 0;
    for (int s = t; s < S_; s += 256) cnt += (action[b * S_ + s] == SENTINEL) ? 1 : 0;
    sh[t] = cnt;
    __syncthreads();
    for (int o = 128; o > 0; o >>= 1) {
        if (t < o) sh[t] += sh[t + o];
        __syncthreads();
    }
    if (t == 0) rowCnt[b] = sh[0];
}

__global__ void scan_rows(const int* __restrict__ rowCnt, int* __restrict__ rowOff) {
    if (threadIdx.x == 0 && blockIdx.x == 0) {
        int acc = 0;
        for (int b = 0; b < B_; ++b) { rowOff[b] = acc; acc += rowCnt[b]; }
    }
}

__global__ void build_occ(const int* __restrict__ action, const int* __restrict__ rowOff,
                          int* __restrict__ occ) {
    const int b = blockIdx.x * blockDim.x + threadIdx.x;
    if (b >= B_) return;
    const int base = rowOff[b];
    int cnt = 0;
    for (int s = 0; s < S_; ++s) {
        cnt += (action[b * S_ + s] == SENTINEL) ? 1 : 0;
        int o = base + cnt - 1;
        o = (o < 0) ? 0 : ((o > B_ - 1) ? B_ - 1 : o);
        occ[b * S_ + s] = o;
    }
}

// ---------------------------------------------------------------------------
// Kernel 3 (the bandwidth kernel): out[r, :] = sentinel ? info[occ[r]] : table[action[r]]
// 262144 rows of 512 f32; 128 float4s per row; NT b128 stores (write-once stream).
// ---------------------------------------------------------------------------
__global__ void scatter_out(const int* __restrict__ action,
                            const int* __restrict__ occ,
                            const float* __restrict__ action_table, // [225,512]
                            const float* __restrict__ info,         // [512,512]
                            float* __restrict__ out)                // [B*S,512]
{
    const int r = blockIdx.x * 2 + (threadIdx.x >> 7); // row in [0, B*S)
    const int q = threadIdx.x & 127;                   // float4 slot in row
    const int a = action[r];
    const float* src = (a == SENTINEL) ? (info + (size_t)occ[r] * D_)
                                       : (action_table + (size_t)a * D_);
    const v4f val = *(const v4f*)(src + q * 4);
    __builtin_nontemporal_store(val, (v4f*)(out + (size_t)r * D_ + q * 4));
}

// ---------------------------------------------------------------------------
extern "C" void kernel_launch(void* const* d_in, const int* in_sizes, int n_in,
                              void* d_out, int out_size, void* d_ws, size_t ws_size,
                              hipStream_t stream) {
    (void)in_sizes; (void)n_in; (void)out_size; (void)ws_size;
    const float* scores     = (const float*)d_in[0];
    const int*   oya        = (const int*)  d_in[1];
    const int*   dora       = (const int*)  d_in[2];
    const float* hrs        = (const float*)d_in[3];
    const int*   action     = (const int*)  d_in[4];
    /* d_in[5] = mask (unused) */
    const float* action_tab = (const float*)d_in[6];
    const float* oya_tab    = (const float*)d_in[7];
    const float* dora_tab   = (const float*)d_in[8];
    const float* ln_gamma   = (const float*)d_in[9];
    const float* ln_beta    = (const float*)d_in[10];
    const float* W_scores   = (const float*)d_in[11];
    const float* b_scores   = (const float*)d_in[12];
    const float* W_hrs      = (const float*)d_in[13];
    const float* b_hrs      = (const float*)d_in[14];
    const float* W_info     = (const float*)d_in[15];
    const float* b_info     = (const float*)d_in[16];
    float* out = (float*)d_out;

    // workspace layout
    char* ws = (char*)d_ws;
    float* concat = (float*)(ws);                                  // 512*384*4 = 786432 B
    float* info   = (float*)(ws + 786432);                         // 512*512*4 = 1048576 B
    int*   occ    = (int*)  (ws + 786432 + 1048576);               // 262144*4  = 1048576 B
    int*   rowCnt = (int*)  (ws + 786432 + 1048576 + 1048576);     // 2048 B
    int*   rowOff = (int*)  (ws + 786432 + 1048576 + 1048576 + 2048);

    // 1) concat features
    build_concat<<<B_, KDIM, 0, stream>>>(scores, oya, dora, hrs, oya_tab, dora_tab,
                                          ln_gamma, ln_beta, W_scores, b_scores,
                                          W_hrs, b_hrs, concat);
    // 2) WMMA GEMM -> embeded_info
    gemm_info<<<256, 128, 0, stream>>>(concat, W_info, b_info, info);
    // 3) sentinel occurrence indices
    count_sentinels<<<B_, 256, 0, stream>>>(action, rowCnt);
    scan_rows<<<1, 32, 0, stream>>>(rowCnt, rowOff);
    build_occ<<<2, 256, 0, stream>>>(action, rowOff, occ);
    // 4) big gather/select/stream-out: (B*S)/2 blocks, 2 rows per 256-thread block
    scatter_out<<<(B_ * S_) / 2, 256, 0, stream>>>(action, occ, action_tab, info, out);
}